// VPLayer_time_21234318311933
// MI455X (gfx1250) — compile-verified
//
#include <hip/hip_runtime.h>
#include <hip/hip_bf16.h>

// Problem constants from the reference: B=4096 rows, N=4097, row fit length NE=4096.
#define TPB   256          // 8 waves (wave32)
#define EPT   16           // elements per thread in the scan phase
#define NE    (TPB * EPT)  // 4096 = N-1
#define PADSZ 4353         // padi(NE) + 1

typedef float v2f __attribute__((ext_vector_type(2)));
typedef float v8f __attribute__((ext_vector_type(8)));

__device__ __forceinline__ int padi(int i) { return i + (i >> 4); }  // bank-conflict pad

#if defined(__gfx1250__) && __has_builtin(__builtin_amdgcn_global_load_async_to_lds_b32)
#define HAVE_ASYNC_LDS 1
typedef __attribute__((address_space(1))) int glb_i;   // global dword (builtin takes int*)
typedef __attribute__((address_space(3))) int lds_i;   // LDS dword
#endif

__global__ __launch_bounds__(TPB)
void vp_rowfit_kernel(const float* __restrict__ x, const float* __restrict__ t,
                      const float* __restrict__ params,
                      float* __restrict__ coeffs_o, float* __restrict__ xhat_o,
                      float* __restrict__ res_o, float* __restrict__ r2_o) {
  __shared__ float s_x[PADSZ];      // staged x row (x[0..NE])
  __shared__ float s_t[PADSZ];      // staged t; reused to hold w = z^r after scan
  __shared__ float s_z[PADSZ];      // z values
  __shared__ float s_scan[TPB];     // block scan workspace (reused for r2 reduce)
  __shared__ float s_part[8][16];   // per-wave reduction partials (cols 0..4 used)
  __shared__ float s_red[16];       // combined sums (WMMA output)
  __shared__ float s_c[2];          // c0, c1

  const int tid = threadIdx.x;
  const int b   = blockIdx.x;
  const float r  = params[0];
  const float ny = params[1];
  const float* xrow = x + (size_t)b * (NE + 1);

  if (tid < 128) ((float*)s_part)[tid] = 0.0f;  // zero unused WMMA columns

  // ---- Stage x row + t into LDS, coalesced (async-to-LDS path on gfx1250) ----
#ifdef HAVE_ASYNC_LDS
  for (int k = 0; k < EPT; ++k) {
    int i = tid + k * TPB;
    __builtin_amdgcn_global_load_async_to_lds_b32((glb_i*)(xrow + i), (lds_i*)&s_x[padi(i)], 0, 0);
    __builtin_amdgcn_global_load_async_to_lds_b32((glb_i*)(t + i),    (lds_i*)&s_t[padi(i)], 0, 0);
  }
  if (tid == 0) {
    __builtin_amdgcn_global_load_async_to_lds_b32((glb_i*)(xrow + NE), (lds_i*)&s_x[padi(NE)], 0, 0);
    __builtin_amdgcn_global_load_async_to_lds_b32((glb_i*)(t + NE),    (lds_i*)&s_t[padi(NE)], 0, 0);
  }
#if __has_builtin(__builtin_amdgcn_s_wait_asynccnt)
  __builtin_amdgcn_s_wait_asynccnt(0);
#else
  asm volatile("s_wait_asynccnt 0" ::: "memory");
#endif
#else
  for (int k = 0; k < EPT; ++k) {
    int i = tid + k * TPB;
    s_x[padi(i)] = xrow[i];
    s_t[padi(i)] = t[i];
  }
  if (tid == 0) { s_x[padi(NE)] = xrow[NE]; s_t[padi(NE)] = t[NE]; }
#endif
  __syncthreads();

  // ---- Trapezoid increments + per-thread serial scan ----
  const int base = tid * EPT;
  float incs[EPT];
  float run = 0.0f;
  for (int j = 0; j < EPT; ++j) {
    int i = base + j;
    float dt  = s_t[padi(i + 1)] - s_t[padi(i)];
    float inc = 0.5f * dt * (s_x[padi(i)] + s_x[padi(i + 1)]);
    incs[j] = inc;
    run += inc;
  }

  // ---- Hillis-Steele block scan over the 256 per-thread totals ----
  s_scan[tid] = run;
  __syncthreads();
  float tot = run;
  for (int off = 1; off < TPB; off <<= 1) {
    float u = (tid >= off) ? s_scan[tid - off] : 0.0f;
    __syncthreads();
    tot += u;
    s_scan[tid] = tot;
    __syncthreads();
  }
  const float excl = tot - run;

  // ---- z, w = z^r, Gram/RHS partial sums; cache z and w in LDS ----
  float g00 = 0.f, g01 = 0.f, g11 = 0.f, d0 = 0.f, d1 = 0.f;
  run = excl;
  for (int j = 0; j < EPT; ++j) {
    int i = base + j;
    run += incs[j];
    float z  = ny + run;
    float w  = powf(z, r);
    s_z[padi(i)] = z;
    s_t[padi(i)] = w;          // t staging is dead now; reuse for w
    float xs = s_x[padi(i + 1)];
    g00 += z * z;  g01 += z * w;  g11 += w * w;
    d0  += xs * z; d1  += xs * w;
  }

  // ---- wave32 reduction of the 5 partials ----
  for (int m = 16; m >= 1; m >>= 1) {
    g00 += __shfl_xor(g00, m, 32);
    g01 += __shfl_xor(g01, m, 32);
    g11 += __shfl_xor(g11, m, 32);
    d0  += __shfl_xor(d0,  m, 32);
    d1  += __shfl_xor(d1,  m, 32);
  }
  if ((tid & 31) == 0) {
    int wv = tid >> 5;
    s_part[wv][0] = g00; s_part[wv][1] = g01; s_part[wv][2] = g11;
    s_part[wv][3] = d0;  s_part[wv][4] = d1;
  }
  __syncthreads();

  // ---- Cross-wave combine: D = ones(16x4) x partials(4x16), two accumulating
  // WMMAs fold all 8 wave partials. Column sums are invariant to the exact
  // K<->(VGPR,half) mapping because A is all-ones. Wave 0 only (EXEC all 1s).
  if (tid < 32) {
#if defined(__gfx1250__) && __has_builtin(__builtin_amdgcn_wmma_f32_16x16x4_f32)
    int q = tid & 15, h = tid >> 4;
    v2f a;  a[0] = 1.0f; a[1] = 1.0f;
    v2f b0; b0[0] = s_part[2 * h + 0][q]; b0[1] = s_part[2 * h + 1][q];
    v2f b1; b1[0] = s_part[2 * h + 4][q]; b1[1] = s_part[2 * h + 5][q];
    v8f acc = {};
    acc = __builtin_amdgcn_wmma_f32_16x16x4_f32(false, a, false, b0, (short)0, acc, false, false);
    acc = __builtin_amdgcn_wmma_f32_16x16x4_f32(false, a, false, b1, (short)0, acc, false, false);
    if (tid < 16) s_red[tid] = acc[0];   // row M=0 = VGPR0, lanes 0..15
#else
    if (tid < 16) {
      float s = 0.0f;
      for (int w = 0; w < 8; ++w) s += s_part[w][tid];
      s_red[tid] = s;
    }
#endif
  }
  __syncthreads();

  // ---- 2x2 Gram inverse -> coeffs ----
  if (tid == 0) {
    float G00 = s_red[0], G01 = s_red[1], G11 = s_red[2];
    float B0 = s_red[3], B1 = s_red[4];
    float inv = 1.0f / (G00 * G11 - G01 * G01);
    float c0 = (B0 * G11 - B1 * G01) * inv;
    float c1 = (B1 * G00 - B0 * G01) * inv;
    s_c[0] = c0; s_c[1] = c1;
    coeffs_o[2 * b + 0] = c0;
    coeffs_o[2 * b + 1] = c1;
  }
  __syncthreads();
  const float c0 = s_c[0], c1 = s_c[1];

  // ---- Residual pass: coalesced float4 stores of x_hat and res ----
  float r2p = 0.0f;
  float4* xh4 = (float4*)(xhat_o + (size_t)b * NE);
  float4* rs4 = (float4*)(res_o  + (size_t)b * NE);
  for (int k = 0; k < 4; ++k) {
    int q = tid + k * TPB;     // float4 index 0..1023
    int m0 = q * 4;
    float hv[4], rv[4];
    for (int e = 0; e < 4; ++e) {
      int m = m0 + e;
      float z  = s_z[padi(m)];
      float w  = s_t[padi(m)];
      float xs = s_x[padi(m + 1)];
      float h  = fmaf(c0, z, c1 * w);
      float re = xs - h;
      hv[e] = h; rv[e] = re;
      r2p += re * re;
    }
    xh4[q] = make_float4(hv[0], hv[1], hv[2], hv[3]);
    rs4[q] = make_float4(rv[0], rv[1], rv[2], rv[3]);
  }

  // ---- r2 reduction ----
  for (int m = 16; m >= 1; m >>= 1) r2p += __shfl_xor(r2p, m, 32);
  if ((tid & 31) == 0) s_scan[tid >> 5] = r2p;
  __syncthreads();
  if (tid == 0) {
    float s = 0.0f;
    for (int w = 0; w < 8; ++w) s += s_scan[w];
    r2_o[b] = s;
  }
}

extern "C" void kernel_launch(void* const* d_in, const int* in_sizes, int n_in,
                              void* d_out, int out_size, void* d_ws, size_t ws_size,
                              hipStream_t stream) {
  (void)n_in; (void)out_size; (void)d_ws; (void)ws_size;
  const float* x      = (const float*)d_in[0];
  const float* t      = (const float*)d_in[1];
  const float* params = (const float*)d_in[2];

  const int Nt = in_sizes[1];        // 4097
  const int B  = in_sizes[0] / Nt;   // 4096
  const int ne = Nt - 1;             // 4096 == NE

  float* out    = (float*)d_out;
  float* coeffs = out;                                // (B,1,2)
  float* xhat   = out + (size_t)B * 2;                // (B,1,NE)
  float* res    = xhat + (size_t)B * ne;              // (B,1,NE)
  float* r2     = res  + (size_t)B * ne;              // (B,1)

  vp_rowfit_kernel<<<B, TPB, 0, stream>>>(x, t, params, coeffs, xhat, res, r2);
}